// PPM_15522011808029
// MI455X (gfx1250) — compile-verified
//
#include <hip/hip_runtime.h>
#include <hip/hip_bf16.h>

typedef __attribute__((ext_vector_type(2))) float v2f;
typedef __attribute__((ext_vector_type(8))) float v8f;

#define E_ 4
#define H_ 64
#define N_ 64
#define B_ 8
#define L_ 8
#define D_ 128
#define DIA_ 110
#define KPD_ 56320ull          // DIA * B * N  (entries per detectorlet)
#define NVOX_ 262144           // H*N*N
#define BN_ 512                // B*N
#define PAD_ 111               // DIA padded for LDS bank-conflict-free access

// ---------------------------------------------------------------------------
// Kernel 1: bilinear rotation of grid_concentration + overwrite slab with xp.
// ---------------------------------------------------------------------------
__global__ void PPM_rotate_kernel(const float* __restrict__ grid,
                                  const float* __restrict__ xp,
                                  const float* __restrict__ theta_ls,
                                  const int* __restrict__ theta_idx,
                                  const int* __restrict__ p_in,
                                  float* __restrict__ rot) {
    int id = blockIdx.x * blockDim.x + threadIdx.x;   // < E*H*N*N
    float theta = theta_ls[theta_idx[0]];
    float ct = cosf(theta), st = sinf(theta);

    int j  = id & 63;
    int i  = (id >> 6) & 63;
    int eh = id >> 12;            // e*H + h
    int h  = eh & 63;
    int e  = eh >> 6;

    const float c = 31.5f;
    float x = (float)j - c;
    float y = (float)i - c;
    float xs = ct * x - st * y + c;
    float ys = st * x + ct * y + c;
    float x0 = floorf(xs), y0 = floorf(ys);
    float wx = xs - x0, wy = ys - y0;
    int ix0 = (int)x0, iy0 = (int)y0;

    const float* img = grid + (size_t)eh * (N_ * N_);
    float acc = 0.0f;
#pragma unroll
    for (int dy = 0; dy < 2; ++dy) {
#pragma unroll
        for (int dx = 0; dx < 2; ++dx) {
            int yi = iy0 + dy;
            int xi = ix0 + dx;
            bool valid = (yi >= 0) & (yi < N_) & (xi >= 0) & (xi < N_);
            float w = (dy ? wy : 1.0f - wy) * (dx ? wx : 1.0f - wx);
            if (valid) acc += img[yi * N_ + xi] * w;
        }
    }

    int r   = h * N_ + i;
    int rel = r - B_ * p_in[0];
    if (rel >= 0 && rel < B_) acc = xp[(e * B_ + rel) * N_ + j];

    rot[id] = acc;
}

// ---------------------------------------------------------------------------
// Kernel 2: attenuation (exclusive cumsum), transmission -> out2, zero out1
// and zero the global SA accumulator. One block, 512 threads.
// ---------------------------------------------------------------------------
__global__ void PPM_atten_kernel(const float* __restrict__ xp,
                                 const float* __restrict__ attCS,
                                 const float* __restrict__ ssc,
                                 const float* __restrict__ pcts,
                                 float* __restrict__ att,
                                 float* __restrict__ sa_glob,
                                 float* __restrict__ out) {
    __shared__ float vals[BN_];
    int t = threadIdx.x;
    int b = t >> 6;
    int j = t & 63;

    float v = 0.0f;
#pragma unroll
    for (int e = 0; e < E_; ++e)
        v += attCS[e] * xp[(e * B_ + b) * N_ + j];
    vals[t] = v;
    __syncthreads();

    float dx = ssc[0] / (float)N_;
    float excl = 0.0f;
    for (int jj = 0; jj < j; ++jj) excl += vals[b * N_ + jj];

    att[t] = __expf(-dx * excl);                      // attenuation_flat
    if (j == 63)
        out[64 + b] = pcts[0] * __expf(-dx * (excl + v));   // out2
    if (t < 64)
        out[t] = 0.0f;                                // out1 init
#pragma unroll
    for (int i = 0; i < L_ * BN_ / BN_; ++i) { }      // (no-op)
    for (int i = t; i < L_ * BN_; i += BN_)
        sa_glob[i] = 0.0f;                            // SA accumulator init
}

// ---------------------------------------------------------------------------
// Kernel 3 (hot): async-stage path data to LDS, per-lane (v,e) gather-FMA,
// one v_wmma_f32_16x16x4_f32 per wave, exp, LDS+global accumulation of SA.
// Grid: 1024 blocks = 32 v-tiles x 32 d-groups; block = 128 threads (4 waves),
// wave w owns d = dgroup*4 + w.
// ---------------------------------------------------------------------------
__global__ void PPM_ray_kernel(const float* __restrict__ conc,     // rot volume
                               const float* __restrict__ FL,       // (E,L)
                               const float* __restrict__ P_len,
                               const int* __restrict__ P_idx,
                               float* __restrict__ sa_glob) {
    __shared__ uint2 stage[4][16][PAD_];   // [wave][v_local][k] = (idx, len)
    __shared__ float lds_sa[L_ * 16];      // exp(-s) partials: [l][v_local]

    const int tid  = threadIdx.x;
    const int wave = tid >> 5;
    const int lane = tid & 31;
    const int hlf  = lane >> 4;            // e-pair select / row-half select
    const int lrow = lane & 15;            // this lane's v_local (and A row)
    const int v_base = (blockIdx.x & 31) * 16;
    const int d      = (blockIdx.x >> 5) * 4 + wave;

    if (tid < L_ * 16) lds_sa[tid] = 0.0f;

    // A-matrix (16x4 f32): A[l][e] = FL[e][l]; rows l>=8 zero.
    // VGPR0 = K=0 (lanes 0-15) / K=2 (lanes 16-31); VGPR1 = K=1 / K=3.
    v2f amat;
    amat.x = (lrow < L_) ? FL[(hlf * 2 + 0) * L_ + lrow] : 0.0f;
    amat.y = (lrow < L_) ? FL[(hlf * 2 + 1) * L_ + lrow] : 0.0f;

    // ---- stage this wave's (d, v-tile) path data into LDS, interleaved ----
    const size_t dbase = (size_t)d * KPD_;
    for (int vl = 0; vl < 16; ++vl) {
        unsigned goff  = (unsigned)((dbase + (size_t)(v_base + vl) * DIA_) * 4u);
        unsigned lbase = (unsigned)(uintptr_t)&stage[wave][vl][0];
        for (int kk = lane; kk < DIA_; kk += 32) {
            unsigned la = lbase + (unsigned)kk * 8u;
            unsigned ga = goff + (unsigned)kk * 4u;
            asm volatile("global_load_async_to_lds_b32 %0, %1, %2"
                         :: "v"(la), "v"(ga), "s"(P_idx) : "memory");
            asm volatile("global_load_async_to_lds_b32 %0, %1, %2"
                         :: "v"(la + 4u), "v"(ga), "s"(P_len) : "memory");
        }
    }
    asm volatile("s_wait_asynccnt 0x0" ::: "memory");

    // ---- per-lane gather-FMA: lane handles (v = lrow, e-pair = 2*hlf) ----
    // Result lands directly in the 4x16 f32 B-matrix layout: no shuffles.
    const float* c0 = conc + (size_t)(hlf * 2 + 0) * NVOX_;
    const float* c1 = conc + (size_t)(hlf * 2 + 1) * NVOX_;
    const uint2* sp = &stage[wave][lrow][0];
    float b0 = 0.0f, b1 = 0.0f;
    for (int kk = 0; kk < DIA_; ++kk) {
        uint2 pr  = sp[kk];                    // ds_load_b64, bank-conflict-free
        int   idx = (int)pr.x;
        float len = __uint_as_float(pr.y);
        b0 += c0[idx] * len;
        b1 += c1[idx] * len;
    }

    v2f bmat; bmat.x = b0; bmat.y = b1;
    v8f cz = {};
    // s[l, v] = sum_e FL[e,l] * per_ray[e,d,v]   (D = A(16x4) x B(4x16))
    v8f s = __builtin_amdgcn_wmma_f32_16x16x4_f32(
        /*neg_a=*/false, amat, /*neg_b=*/false, bmat,
        /*c_mod=*/(short)0, cz, /*reuse_a=*/false, /*reuse_b=*/false);

    // valid rows l=0..7 live in lanes 0-15 (VGPR r -> M=r); accumulate exp(-s)
    if (hlf == 0) {
#pragma unroll
        for (int r = 0; r < 8; ++r)
            atomicAdd(&lds_sa[r * 16 + lrow], __expf(-s[r]));
    }
    __syncthreads();

    // one global add per (l, v) per block into the SA accumulator
    if (tid < L_ * 16) {
        int l = tid >> 4;
        int n = tid & 15;
        atomicAdd(&sa_glob[l * BN_ + v_base + n], lds_sa[tid]);
    }
}

// ---------------------------------------------------------------------------
// Kernel 4: sig = probe_cts * attenuation * fl_map * mean_d(exp(-s));
// reduce over j into out1[l,b].  4096 threads.
// ---------------------------------------------------------------------------
__global__ void PPM_final_kernel(const float* __restrict__ sa_glob,
                                 const float* __restrict__ att,
                                 const float* __restrict__ xp,
                                 const float* __restrict__ dfu,
                                 const int* __restrict__ l2e,
                                 const float* __restrict__ pcts,
                                 float* __restrict__ out) {
    int t = blockIdx.x * blockDim.x + threadIdx.x;    // < L*BN = 4096
    int l = t >> 9;
    int v = t & (BN_ - 1);
    int b = v >> 6;
    int j = v & 63;
    float sam = sa_glob[t] * (1.0f / (float)D_);
    float fl  = xp[(l2e[l] * B_ + b) * N_ + j] * dfu[l];
    float sig = pcts[0] * att[v] * fl * sam;
    atomicAdd(&out[l * B_ + b], sig);
}

// ---------------------------------------------------------------------------
extern "C" void kernel_launch(void* const* d_in, const int* in_sizes, int n_in,
                              void* d_out, int out_size, void* d_ws, size_t ws_size,
                              hipStream_t stream) {
    const float* grid   = (const float*)d_in[0];   // (E,H,N,N)
    const float* xp     = (const float*)d_in[1];   // (E,B,N)
    const float* thetas = (const float*)d_in[2];   // (16,)
    const float* attCS  = (const float*)d_in[3];   // (E,)
    const float* FL     = (const float*)d_in[4];   // (E,L)
    const float* dfu    = (const float*)d_in[5];   // (L,)
    const float* P_len  = (const float*)d_in[6];   // (D,K)
    const float* ssc    = (const float*)d_in[7];   // scalar
    const float* pcts   = (const float*)d_in[8];   // scalar
    const int*   l2e    = (const int*)d_in[9];     // (L,)
    const int*   P_idx  = (const int*)d_in[10];    // (D,K)
    const int*   tidx   = (const int*)d_in[11];    // scalar
    const int*   p_in   = (const int*)d_in[12];    // scalar

    float* out     = (float*)d_out;                // [0:64)=out1, [64:72)=out2
    float* rot     = (float*)d_ws;                 // E*NVOX floats = 4 MB
    float* att     = rot + (size_t)E_ * NVOX_;     // 512 floats
    float* sa_glob = att + BN_;                    // L*BN = 4096 floats

    PPM_rotate_kernel<<<(E_ * H_ * N_ * N_) / 256, 256, 0, stream>>>(
        grid, xp, thetas, tidx, p_in, rot);

    PPM_atten_kernel<<<1, BN_, 0, stream>>>(xp, attCS, ssc, pcts, att, sa_glob, out);

    PPM_ray_kernel<<<32 * 32, 128, 0, stream>>>(rot, FL, P_len, P_idx, sa_glob);

    PPM_final_kernel<<<(L_ * BN_) / 256, 256, 0, stream>>>(
        sa_glob, att, xp, dfu, l2e, pcts, out);
}